// TemporalComplementarityAttention_77592879169739
// MI455X (gfx1250) — compile-verified
//
#include <hip/hip_runtime.h>
#include <hip/hip_bf16.h>
#include <math.h>

// Shapes (fixed by the reference): B=4, N=1024, D=128, H=64
#define BATCH 4
#define NTOK  1024
#define DIM   128
#define HID   64

typedef __attribute__((ext_vector_type(2))) float v2f;
typedef __attribute__((ext_vector_type(4))) float f4;
typedef __attribute__((ext_vector_type(8))) float v8f;

// Hardware tanh (TRANS32 op on CDNA5). Falls back to inline asm if the
// builtin is not declared by this toolchain.
__device__ __forceinline__ float fast_tanh(float x) {
#if __has_builtin(__builtin_amdgcn_tanhf)
  return __builtin_amdgcn_tanhf(x);
#else
  float r;
  asm volatile("v_tanh_f32 %0, %1" : "=v"(r) : "v"(x));
  return r;
#endif
}

// -----------------------------------------------------------------------------
// Kernel 1: hi' = E @ Wa + b1  and  hj = E @ Wb, via V_WMMA_F32_16X16X4_F32.
// E is (B*N, D) = (4096, 128) f32, W1 is (2D, H) = (256, 64) f32 row-major.
// grid = (M/16=256, H/16=4, 2 halves), block = 32 (one wave -> one 16x16 tile).
//
// A-matrix 16x4 f32 layout (ISA 7.12.2): lanes 0-15 hold M=lane; v0 = K=0
// (lanes 0-15) / K=2 (lanes 16-31); v1 = K=1 / K=3.
// B-matrix 4x16 f32: rows striped across lanes per VGPR, halves split like C:
// v0 = K=0 (lanes 0-15) / K=2 (lanes 16-31); v1 = K=1 / K=3.
// C/D 16x16 f32: VGPR r -> M=r (lanes 0-15), M=r+8 (lanes 16-31), N = lane&15.
// -----------------------------------------------------------------------------
__global__ void __launch_bounds__(32)
proj_kernel(const float* __restrict__ E, const float* __restrict__ W1,
            const float* __restrict__ b1, float* __restrict__ Hbuf) {
  const int lane = threadIdx.x;            // 0..31, EXEC all-ones (no divergence)
  const int m0   = blockIdx.x * 16;        // row tile in flattened (B*N)
  const int n0   = blockIdx.y * 16;        // column tile in H
  const int half = blockIdx.z;             // 0 = Wa (-> hi'), 1 = Wb (-> hj)

  const float* W = W1 + half * (DIM * HID);    // rows [half*128, half*128+128)
  const int row  = m0 + (lane & 15);
  const int koff = (lane >> 4) * 2;            // 0 for lanes 0-15, 2 for 16-31
  const int ncol = n0 + (lane & 15);

  v8f c = {};
  #pragma unroll
  for (int k = 0; k < DIM; k += 4) {
    v2f a, b;
    a.x = E[row * DIM + (k + koff)];
    a.y = E[row * DIM + (k + koff + 1)];
    b.x = W[(k + koff) * HID + ncol];
    b.y = W[(k + koff + 1) * HID + ncol];
    c = __builtin_amdgcn_wmma_f32_16x16x4_f32(
        /*neg_a=*/false, a, /*neg_b=*/false, b,
        /*c_mod=*/(short)0, c, /*reuse_a=*/false, /*reuse_b=*/false);
  }

  // Fold b1 into hi' so the pairwise stage computes relu(hi'[i]+hj[j]).
  float* out = Hbuf + (size_t)half * (BATCH * NTOK * HID);
  const float bias  = (half == 0) ? b1[ncol] : 0.0f;
  const int   rbase = m0 + ((lane >> 4) * 8);
  #pragma unroll
  for (int r = 0; r < 8; ++r) {
    out[(size_t)(rbase + r) * HID + ncol] = c[r] + bias;
  }
}

// -----------------------------------------------------------------------------
// Kernel 2: pairwise stage, upper-triangular 32x32 tiles, both directions.
//   d_ij = sum_h relu(hi'[i,h] + hj[j,h]) * w2[h]
//   out[b,i,j] = out[b,j,i] = -0.5*(tanh(d_ij + b2) + tanh(d_ji + b2))
// Each thread owns a 2x2 (i,j) sub-tile -> 8 accumulators. The mirrored
// (j,i) tile is staged in LDS and drained with coalesced row writes.
// -----------------------------------------------------------------------------
__device__ __forceinline__ float relu_dot4(f4 a, f4 b, f4 w, float acc) {
  f4 s = a + b;
  acc = fmaf(fmaxf(s.x, 0.0f), w.x, acc);
  acc = fmaf(fmaxf(s.y, 0.0f), w.y, acc);
  acc = fmaf(fmaxf(s.z, 0.0f), w.z, acc);
  acc = fmaf(fmaxf(s.w, 0.0f), w.w, acc);
  return acc;
}

__global__ void __launch_bounds__(256)
pair_kernel(const float* __restrict__ Hi, const float* __restrict__ Hj,
            const float* __restrict__ W2, const float* __restrict__ b2p,
            float* __restrict__ out) {
  const int ti = blockIdx.x, tj = blockIdx.y, b = blockIdx.z;
  if (tj < ti) return;                       // upper triangle only (uniform exit)

  __shared__ float sHiA[32 * HID];           // hi' rows of i-block
  __shared__ float sHiB[32 * HID];           // hi' rows of j-block
  __shared__ float sHjA[32 * HID];           // hj  rows of i-block
  __shared__ float sHjB[32 * HID];           // hj  rows of j-block
  __shared__ float sW2[HID];
  __shared__ float sT[32 * 32];              // transposed result tile

  const int tid = threadIdx.x;
  const int i0 = ti * 32, j0 = tj * 32;
  const float* HiBb = Hi + (size_t)b * NTOK * HID;
  const float* HjBb = Hj + (size_t)b * NTOK * HID;
  const f4* gHiA = (const f4*)(HiBb + (size_t)i0 * HID);
  const f4* gHiB = (const f4*)(HiBb + (size_t)j0 * HID);
  const f4* gHjA = (const f4*)(HjBb + (size_t)i0 * HID);
  const f4* gHjB = (const f4*)(HjBb + (size_t)j0 * HID);

  #pragma unroll
  for (int idx = tid; idx < 32 * HID / 4; idx += 256) {  // 512 float4 each
    ((f4*)sHiA)[idx] = gHiA[idx];
    ((f4*)sHiB)[idx] = gHiB[idx];
    ((f4*)sHjA)[idx] = gHjA[idx];
    ((f4*)sHjB)[idx] = gHjB[idx];
  }
  if (tid < HID / 4) ((f4*)sW2)[tid] = ((const f4*)W2)[tid];
  __syncthreads();

  const int ii0 = (tid >> 4) * 2;            // 0..30 within i-block
  const int jj0 = (tid & 15) * 2;            // 0..30 within j-block

  float dij[2][2] = {{0.f, 0.f}, {0.f, 0.f}};
  float dji[2][2] = {{0.f, 0.f}, {0.f, 0.f}};

  #pragma unroll
  for (int h = 0; h < HID; h += 4) {
    const f4 w    = *(const f4*)&sW2[h];
    const f4 hiA0 = *(const f4*)&sHiA[(ii0 + 0) * HID + h];
    const f4 hiA1 = *(const f4*)&sHiA[(ii0 + 1) * HID + h];
    const f4 hjA0 = *(const f4*)&sHjA[(ii0 + 0) * HID + h];
    const f4 hjA1 = *(const f4*)&sHjA[(ii0 + 1) * HID + h];
    const f4 hiB0 = *(const f4*)&sHiB[(jj0 + 0) * HID + h];
    const f4 hiB1 = *(const f4*)&sHiB[(jj0 + 1) * HID + h];
    const f4 hjB0 = *(const f4*)&sHjB[(jj0 + 0) * HID + h];
    const f4 hjB1 = *(const f4*)&sHjB[(jj0 + 1) * HID + h];
    // forward: d_ij uses hi'[i] + hj[j]
    dij[0][0] = relu_dot4(hiA0, hjB0, w, dij[0][0]);
    dij[0][1] = relu_dot4(hiA0, hjB1, w, dij[0][1]);
    dij[1][0] = relu_dot4(hiA1, hjB0, w, dij[1][0]);
    dij[1][1] = relu_dot4(hiA1, hjB1, w, dij[1][1]);
    // backward: d_ji uses hi'[j] + hj[i]
    dji[0][0] = relu_dot4(hiB0, hjA0, w, dji[0][0]);
    dji[0][1] = relu_dot4(hiB1, hjA0, w, dji[0][1]);
    dji[1][0] = relu_dot4(hiB0, hjA1, w, dji[1][0]);
    dji[1][1] = relu_dot4(hiB1, hjA1, w, dji[1][1]);
  }

  const float b2v = b2p[0];
  #pragma unroll
  for (int u = 0; u < 2; ++u) {
    #pragma unroll
    for (int v = 0; v < 2; ++v) {
      const int gi = i0 + ii0 + u;
      const int gj = j0 + jj0 + v;
      const float s =
          -0.5f * (fast_tanh(dij[u][v] + b2v) + fast_tanh(dji[u][v] + b2v));
      out[((size_t)b * NTOK + gi) * NTOK + gj] = s;   // direct tile: coalesced
      sT[(jj0 + v) * 32 + (ii0 + u)] = s;             // stage mirror tile
    }
  }
  __syncthreads();

  // Drain transposed tile with coalesced row writes (diag tiles rewrite the
  // same values -> benign).
  #pragma unroll
  for (int idx = tid; idx < 32 * 32; idx += 256) {
    const int r = idx >> 5, c = idx & 31;
    out[((size_t)b * NTOK + (j0 + r)) * NTOK + (i0 + c)] = sT[idx];
  }
}

extern "C" void kernel_launch(void* const* d_in, const int* in_sizes, int n_in,
                              void* d_out, int out_size, void* d_ws, size_t ws_size,
                              hipStream_t stream) {
  (void)in_sizes; (void)n_in; (void)out_size; (void)ws_size;
  const float* E  = (const float*)d_in[0];   // (4,1024,128)
  const float* W1 = (const float*)d_in[1];   // (256,64)
  const float* b1 = (const float*)d_in[2];   // (64,)
  const float* W2 = (const float*)d_in[3];   // (64,1)
  const float* b2 = (const float*)d_in[4];   // (1,)
  float* out  = (float*)d_out;               // (4,1024,1024)
  float* Hbuf = (float*)d_ws;                // hi' then hj, each 4096*64 f32 (2 MB)

  proj_kernel<<<dim3((BATCH * NTOK) / 16, HID / 16, 2), 32, 0, stream>>>(E, W1, b1, Hbuf);
  pair_kernel<<<dim3(NTOK / 32, NTOK / 32, BATCH), 256, 0, stream>>>(
      Hbuf, Hbuf + (size_t)BATCH * NTOK * HID, W2, b2, out);
}